// EdgeStructureBranch_11613591568829
// MI455X (gfx1250) — compile-verified
//
#include <hip/hip_runtime.h>
#include <hip/hip_bf16.h>

typedef float v2f __attribute__((ext_vector_type(2)));
typedef float v8f __attribute__((ext_vector_type(8)));

constexpr int IH = 512;
constexpr int IW = 512;
constexpr int NB = 64;
constexpr int HW = IH * IW;
constexpr int WPR = 16;            // 32-bit words per row
constexpr int WPI = IH * WPR;      // words per image = 8192

__device__ __forceinline__ int reflect101(int i) {
    if (i < 0) i = -i;
    if (i >= IH) i = 2 * IH - 2 - i;
    return i;
}

// ---------------------------------------------------------------------------
// Kernel 1: fused gray + Sobel/Laplace + NMS + per-tile stats.
// grid (16,16,64): (word-col tile, row tile, image). block (32,8): lane = x.
// Outputs: weak/strong bitmasks (1 bit/pixel), per-tile partial sums.
// ---------------------------------------------------------------------------
__global__ __launch_bounds__(256)
void fused_stencil_kernel(const float* __restrict__ x,
                          unsigned* __restrict__ weakW,
                          unsigned* __restrict__ strongW,
                          float* __restrict__ partials) {
    __shared__ float G[36 * 36];   // gray tile, halo 2 (reflect-101 vs image)
    __shared__ float Mg[34 * 34];  // |gx|+|gy| tile, halo 1 (zero outside image)
    __shared__ float red[256];

    const int img = blockIdx.z;
    const int bx = blockIdx.x;          // 0..15 word column
    const int by = blockIdx.y;          // 0..15 row tile
    const int tx = threadIdx.x;         // 0..31 (one wave per ty row)
    const int ty = threadIdx.y;         // 0..7
    const int tid = ty * 32 + tx;
    const int x0 = bx * 32, y0 = by * 32;

    const float* xr = x + (size_t)img * 3 * HW;
    const float* xg = xr + HW;
    const float* xb = xr + 2 * HW;

    // Load gray with reflect-101 halo of 2.
    for (int idx = tid; idx < 36 * 36; idx += 256) {
        int yy = idx / 36, xx = idx % 36;
        int iy = reflect101(y0 - 2 + yy);
        int ix = reflect101(x0 - 2 + xx);
        int o = iy * IW + ix;
        float r8 = rintf(fminf(fmaxf(xr[o], 0.f), 1.f) * 255.f);
        float g8 = rintf(fminf(fmaxf(xg[o], 0.f), 1.f) * 255.f);
        float b8 = rintf(fminf(fmaxf(xb[o], 0.f), 1.f) * 255.f);
        G[idx] = rintf(0.299f * r8 + 0.587f * g8 + 0.114f * b8);
    }
    __syncthreads();

    // L1 gradient magnitude with halo 1; zero outside the image (NMS shift is
    // zero-padded in the reference).
    for (int idx = tid; idx < 34 * 34; idx += 256) {
        int my = idx / 34, mx = idx % 34;
        int iy = y0 - 1 + my, ix = x0 - 1 + mx;
        float m = 0.f;
        if (iy >= 0 && iy < IH && ix >= 0 && ix < IW) {
            const float* gp = &G[my * 36 + mx];  // top-left of 3x3
            float gxv = -gp[0] + gp[2] - 2.f * gp[36] + 2.f * gp[38] - gp[72] + gp[74];
            float gyv = -gp[0] - 2.f * gp[1] - gp[2] + gp[72] + 2.f * gp[73] + gp[74];
            m = fabsf(gxv) + fabsf(gyv);
        }
        Mg[idx] = m;
    }
    __syncthreads();

    float sS = 0.f, sA = 0.f, sA2 = 0.f, sL = 0.f, sL2 = 0.f;

    #pragma unroll
    for (int j = 0; j < 4; ++j) {
        int yl = ty + 8 * j;  // 0..31
        const float* gp = &G[(yl + 1) * 36 + (tx + 1)];  // top-left of 3x3 around center
        float c   = gp[37];
        float gxv = -gp[0] + gp[2] - 2.f * gp[36] + 2.f * gp[38] - gp[72] + gp[74];
        float gyv = -gp[0] - 2.f * gp[1] - gp[2] + gp[72] + 2.f * gp[73] + gp[74];
        float lap = gp[1] + gp[36] - 4.f * c + gp[38] + gp[73];
        float st  = sqrtf(gxv * gxv + gyv * gyv);
        float an  = atan2f(gyv, gxv);
        sS += st; sA += an; sA2 += an * an; sL += lap; sL2 += lap * lap;

        // direction sector in degrees mod 180
        float a = an * 57.29577951308232f;
        a = fmodf(a, 180.f);
        if (a < 0.f) a += 180.f;

        const float* mp = &Mg[(yl + 1) * 34 + (tx + 1)];
        float mc = mp[0];
        float n1, n2;
        if (a < 22.5f || a >= 157.5f) { n1 = mp[1];        n2 = mp[-1]; }
        else if (a < 67.5f)           { n1 = mp[-34 + 1];  n2 = mp[34 - 1]; }
        else if (a < 112.5f)          { n1 = mp[-34];      n2 = mp[34]; }
        else                          { n1 = mp[-34 - 1];  n2 = mp[34 + 1]; }
        float nms = (mc >= n1 && mc >= n2) ? mc : 0.f;

        bool weakp   = nms >= 50.f;
        bool strongp = nms >= 150.f;
        unsigned wwd = (unsigned)__ballot(weakp);   // wave32: low 32 bits
        unsigned swd = (unsigned)__ballot(strongp);
        if (tx == 0) {
            int row = y0 + yl;
            size_t wo = (size_t)img * WPI + (size_t)row * WPR + bx;
            weakW[wo] = wwd;
            strongW[wo] = swd;
        }
    }

    // Deterministic per-block reduction of the 5 stats.
    float vals[5] = {sS, sA, sA2, sL, sL2};
    for (int s = 0; s < 5; ++s) {
        red[tid] = vals[s];
        __syncthreads();
        for (int off = 128; off > 0; off >>= 1) {
            if (tid < off) red[tid] += red[tid + off];
            __syncthreads();
        }
        if (tid == 0)
            partials[((size_t)img * 256 + by * 16 + bx) * 5 + s] = red[0];
        __syncthreads();
    }
}

// ---------------------------------------------------------------------------
// Kernel 2: hysteresis flood fill, fully LDS-resident, one workgroup/image.
// 16 iterations of edges = weak & dilate3(edges), then popcounts.
// ---------------------------------------------------------------------------
__global__ __launch_bounds__(512)
void hysteresis_kernel(const unsigned* __restrict__ weakW,
                       const unsigned* __restrict__ strongW,
                       unsigned* __restrict__ counts) {
    __shared__ unsigned Wm[WPI];
    __shared__ unsigned Sm[WPI];
    __shared__ unsigned red[512];
    const int img = blockIdx.x;
    const int t = threadIdx.x;

    for (int j = 0; j < 16; ++j) {
        int idx = t + 512 * j;
        Wm[idx] = weakW[(size_t)img * WPI + idx];
        Sm[idx] = strongW[(size_t)img * WPI + idx];
    }
    __syncthreads();

    const int c = t & 15;      // word column
    const int r0 = t >> 4;     // 0..31, rows r0 + 32*j
    unsigned nw[16];

    for (int it = 0; it < 16; ++it) {
        for (int j = 0; j < 16; ++j) {
            int r = r0 + 32 * j;
            unsigned acc = 0;
            #pragma unroll
            for (int dr = -1; dr <= 1; ++dr) {
                int rr = r + dr;
                if (rr < 0 || rr >= IH) continue;
                unsigned s = Sm[rr * 16 + c];
                unsigned l = (c > 0)  ? Sm[rr * 16 + c - 1] : 0u;
                unsigned rt = (c < 15) ? Sm[rr * 16 + c + 1] : 0u;
                acc |= s | (s << 1) | (s >> 1) | (l >> 31) | (rt << 31);
            }
            nw[j] = acc & Wm[r * 16 + c];
        }
        __syncthreads();
        for (int j = 0; j < 16; ++j) Sm[(r0 + 32 * j) * 16 + c] = nw[j];
        __syncthreads();
    }

    // edge count + dilated-edge count
    unsigned ec = 0, dc = 0;
    for (int j = 0; j < 16; ++j) {
        int r = r0 + 32 * j;
        ec += __popc(Sm[r * 16 + c]);
        unsigned acc = 0;
        #pragma unroll
        for (int dr = -1; dr <= 1; ++dr) {
            int rr = r + dr;
            if (rr < 0 || rr >= IH) continue;
            unsigned s = Sm[rr * 16 + c];
            unsigned l = (c > 0)  ? Sm[rr * 16 + c - 1] : 0u;
            unsigned rt = (c < 15) ? Sm[rr * 16 + c + 1] : 0u;
            acc |= s | (s << 1) | (s >> 1) | (l >> 31) | (rt << 31);
        }
        dc += __popc(acc);
    }
    red[t] = ec; __syncthreads();
    for (int off = 256; off > 0; off >>= 1) { if (t < off) red[t] += red[t + off]; __syncthreads(); }
    if (t == 0) counts[img * 2] = red[0];
    __syncthreads();
    red[t] = dc; __syncthreads();
    for (int off = 256; off > 0; off >>= 1) { if (t < off) red[t] += red[t + off]; __syncthreads(); }
    if (t == 0) counts[img * 2 + 1] = red[0];
}

// ---------------------------------------------------------------------------
// Kernel 3: assemble 5 features/image, then the 2-layer MLP with f32 WMMA
// (V_WMMA_F32_16X16X4_F32). One block, 8 waves.
// GEMM1: [64 x 8pad] x [8pad x 32] -> 8 tiles, 1/wave.
// GEMM2: [64 x 32]   x [32 x 64]   -> 16 tiles, 2/wave.
// ---------------------------------------------------------------------------
__global__ __launch_bounds__(256)
void head_kernel(const float* __restrict__ partials,
                 const unsigned* __restrict__ counts,
                 const float* __restrict__ W1, const float* __restrict__ b1,
                 const float* __restrict__ W2, const float* __restrict__ b2,
                 float* __restrict__ out) {
    __shared__ float F[64][8];    // features, K padded 5 -> 8
    __shared__ float Hl[64][32];  // hidden activations
    const int tid = threadIdx.x;

    if (tid < 64) {
        float s0 = 0.f, s1 = 0.f, s2 = 0.f, s3 = 0.f, s4 = 0.f;
        const float* p = partials + (size_t)tid * 256 * 5;
        for (int b = 0; b < 256; ++b) {
            s0 += p[b * 5 + 0]; s1 += p[b * 5 + 1]; s2 += p[b * 5 + 2];
            s3 += p[b * 5 + 3]; s4 += p[b * 5 + 4];
        }
        const float N = 262144.f;
        float ecnt = (float)counts[tid * 2];
        float dcnt = (float)counts[tid * 2 + 1];
        float mA = s1 / N, mL = s3 / N;
        F[tid][0] = ecnt / N;                        // edge_density
        F[tid][1] = s0 / N / 255.f;                  // mean_edge_strength
        F[tid][2] = dcnt / (ecnt + 1e-10f);          // continuity
        F[tid][3] = s2 / N - mA * mA;                // angle_var
        F[tid][4] = (s4 / N - mL * mL) / 1000.f;     // high_freq
        F[tid][5] = 0.f; F[tid][6] = 0.f; F[tid][7] = 0.f;
    }
    __syncthreads();

    const int wave = tid >> 5, lane = tid & 31;
    const int lh = lane >> 4;   // lane half selects K pair / M+8
    const int ll = lane & 15;   // M row for A, N col for B/C/D

    // ---- GEMM1: h = relu(F @ W1 + b1) ----
    {
        int m0 = (wave >> 1) * 16, n0 = (wave & 1) * 16;
        v8f acc = {};
        #pragma unroll
        for (int k0 = 0; k0 < 8; k0 += 4) {
            int ka = k0 + 2 * lh;
            v2f a, b;
            a.x = F[m0 + ll][ka];
            a.y = F[m0 + ll][ka + 1];
            b.x = (ka < 5)     ? W1[ka * 32 + n0 + ll]       : 0.f;
            b.y = (ka + 1 < 5) ? W1[(ka + 1) * 32 + n0 + ll] : 0.f;
            acc = __builtin_amdgcn_wmma_f32_16x16x4_f32(
                false, a, false, b, (short)0, acc, false, false);
        }
        #pragma unroll
        for (int v = 0; v < 8; ++v) {
            int m = m0 + v + 8 * lh, n = n0 + ll;
            Hl[m][n] = fmaxf(acc[v] + b1[n], 0.f);
        }
    }
    __syncthreads();

    // ---- GEMM2: out = relu(h @ W2 + b2) ----
    #pragma unroll
    for (int rep = 0; rep < 2; ++rep) {
        int tIdx = wave + 8 * rep;
        int m0 = (tIdx >> 2) * 16, n0 = (tIdx & 3) * 16;
        v8f acc = {};
        #pragma unroll
        for (int k0 = 0; k0 < 32; k0 += 4) {
            int ka = k0 + 2 * lh;
            v2f a, b;
            a.x = Hl[m0 + ll][ka];
            a.y = Hl[m0 + ll][ka + 1];
            b.x = W2[ka * 64 + n0 + ll];
            b.y = W2[(ka + 1) * 64 + n0 + ll];
            acc = __builtin_amdgcn_wmma_f32_16x16x4_f32(
                false, a, false, b, (short)0, acc, false, false);
        }
        #pragma unroll
        for (int v = 0; v < 8; ++v) {
            int m = m0 + v + 8 * lh, n = n0 + ll;
            out[m * 64 + n] = fmaxf(acc[v] + b2[n], 0.f);
        }
    }
}

// ---------------------------------------------------------------------------
extern "C" void kernel_launch(void* const* d_in, const int* in_sizes, int n_in,
                              void* d_out, int out_size, void* d_ws, size_t ws_size,
                              hipStream_t stream) {
    const float* x  = (const float*)d_in[0];
    const float* W1 = (const float*)d_in[1];
    const float* b1 = (const float*)d_in[2];
    const float* W2 = (const float*)d_in[3];
    const float* b2 = (const float*)d_in[4];
    float* out = (float*)d_out;

    char* ws = (char*)d_ws;
    size_t off = 0;
    unsigned* weakW   = (unsigned*)(ws + off); off += (size_t)NB * WPI * 4;   // 2 MB
    unsigned* strongW = (unsigned*)(ws + off); off += (size_t)NB * WPI * 4;   // 2 MB
    float*    partials = (float*)(ws + off);   off += (size_t)NB * 256 * 5 * 4;
    unsigned* counts  = (unsigned*)(ws + off); off += (size_t)NB * 2 * 4;

    dim3 g1(16, 16, NB), blk1(32, 8);
    fused_stencil_kernel<<<g1, blk1, 0, stream>>>(x, weakW, strongW, partials);
    hysteresis_kernel<<<NB, 512, 0, stream>>>(weakW, strongW, counts);
    head_kernel<<<1, 256, 0, stream>>>(partials, counts, W1, b1, W2, b2, out);
}